// DGCNN_86268713107606
// MI455X (gfx1250) — compile-verified
//
#include <hip/hip_runtime.h>
#include <cstddef>

// ---------------------------------------------------------------------------
// DGCNN forward for MI455X (gfx1250, wave32).
// WMMA (V_WMMA_F32_16X16X4_F32) used for the two dense GEMM stages:
//   * pairwise-distance gram per frame (2048x2048, K=9/32), LDS-staged tiles
//   * lin1 (96->1024) fused with max-pool over points
// Everything else (top-k, edge MLP chain with global BN, output MLP) is
// VALU with deterministic reductions (no float atomic adds anywhere).
// ---------------------------------------------------------------------------

typedef __attribute__((ext_vector_type(2))) float v2f;
typedef __attribute__((ext_vector_type(8))) float v8f;

#define NPTS 2048
#define NFRM 32
#define KNN  20

static __device__ __forceinline__ v8f wmma_f32_k4(v2f a, v2f b, v8f c) {
  // D = A(16x4) * B(4x16) + C, fp32, wave32.
  return __builtin_amdgcn_wmma_f32_16x16x4_f32(false, a, false, b,
                                               (short)0, c, false, false);
}

// --------------------------- squared norms ---------------------------------
__global__ __launch_bounds__(256) void sqnorm_kernel(const float* __restrict__ X,
                                                     float* __restrict__ sq, int C) {
  int p = blockIdx.x * 256 + threadIdx.x;          // 0 .. NFRM*NPTS-1
  float s = 0.f;
  for (int c = 0; c < C; ++c) {
    float v = X[(size_t)p * C + c];
    s += v * v;
  }
  sq[p] = s;
}

// ------------------- distance gram tile via fp32 WMMA ----------------------
// One wave computes a 16x16 tile of D[f] = sqi + sqj - 2 * X X^T.
// C is compile-time; A/B tiles staged in LDS, K zero-padded to KP (mult of 4)
// so the WMMA feed is branch-free ds_load_b64 pairs.
template<int C, int KP>
__global__ __launch_bounds__(32) void dist_wmma(const float* __restrict__ X,
                                                const float* __restrict__ sq,
                                                float* __restrict__ D, int f) {
  __shared__ float sA[16 * KP];
  __shared__ float sB[16 * KP];
  int i0 = blockIdx.x * 16, j0 = blockIdx.y * 16;
  int l = threadIdx.x;
  int half = l >> 4;        // 0: K=k0,k0+1   1: K=k0+2,k0+3
  int m = l & 15;           // row (A) / col (B) index within tile
  const float* Xf  = X  + (size_t)f * NPTS * C;
  const float* sqf = sq + (size_t)f * NPTS;

  if (KP > C) {             // compile-time: zero the K padding once
    for (int i = l; i < 16 * KP; i += 32) { sA[i] = 0.f; sB[i] = 0.f; }
    __syncthreads();
  }
#pragma unroll
  for (int i = l; i < 16 * C; i += 32) {
    int r = i / C, c = i % C;
    sA[r * KP + c] = Xf[(size_t)(i0 + r) * C + c];
    sB[r * KP + c] = Xf[(size_t)(j0 + r) * C + c];
  }
  __syncthreads();

  v8f acc = {};
#pragma unroll
  for (int k0 = 0; k0 < KP; k0 += 4) {
    int ka = k0 + 2 * half;
    v2f a = *(const v2f*)&sA[m * KP + ka];   // KP, ka even -> 8B aligned
    v2f b = *(const v2f*)&sB[m * KP + ka];
    acc = wmma_f32_k4(a, b, acc);
  }

  float sj = sqf[j0 + m];
#pragma unroll
  for (int r = 0; r < 8; ++r) {
    int Mi = r + 8 * half;
    D[(size_t)(i0 + Mi) * NPTS + (j0 + m)] = sqf[i0 + Mi] + sj - 2.0f * acc[r];
  }
}

// ------------------------------ top-k --------------------------------------
// One block per row; K iterations of deterministic argmin (lowest-index tie
// break, matching jax.lax.top_k on -d).
__global__ __launch_bounds__(256) void topk_kernel(const float* __restrict__ D,
                                                   int* __restrict__ nb, int f) {
  __shared__ float row[NPTS];
  __shared__ float rv[256];
  __shared__ int   ri[256];
  int n = blockIdx.x, tid = threadIdx.x;
  const float* Dr = D + (size_t)n * NPTS;
  for (int j = tid; j < NPTS; j += 256) row[j] = Dr[j];
  __syncthreads();
  int* out = nb + ((size_t)f * NPTS + n) * KNN;
  for (int it = 0; it < KNN; ++it) {
    float bv = __builtin_inff();
    int bi = NPTS;
    for (int j = tid; j < NPTS; j += 256) {
      float v = row[j];
      if (v < bv) { bv = v; bi = j; }
    }
    rv[tid] = bv; ri[tid] = bi;
    __syncthreads();
    for (int s = 128; s > 0; s >>= 1) {
      if (tid < s) {
        float v2 = rv[tid + s]; int i2 = ri[tid + s];
        if (v2 < rv[tid] || (v2 == rv[tid] && i2 < ri[tid])) { rv[tid] = v2; ri[tid] = i2; }
      }
      __syncthreads();
    }
    if (tid == 0) { out[it] = ri[0]; row[ri[0]] = __builtin_inff(); }
    __syncthreads();
  }
}

// ----------------------- edge MLP chain (3 stages) -------------------------
// STAGE 0: h0 raw -> partial (sum, sumsq) per channel
// STAGE 1: h0->BN0->relu->h1 raw -> partial stats
// STAGE 2: full chain -> max over K -> feats
template<int CIN, int STAGE>
__global__ __launch_bounds__(256) void edge_chain(
    const float* __restrict__ X, const int* __restrict__ nb,
    const float* __restrict__ W0, const float* __restrict__ b0,
    const float* __restrict__ sc0, const float* __restrict__ sh0,
    const float* __restrict__ W1, const float* __restrict__ b1,
    const float* __restrict__ sc1, const float* __restrict__ sh1,
    const float* __restrict__ W2, const float* __restrict__ b2,
    float* __restrict__ part, float* __restrict__ outFeat) {
  constexpr int CE = 2 * CIN;
  __shared__ float sW0[32 * CE];
  __shared__ float sW1[32 * 32];
  __shared__ float sW2[32 * 32];
  __shared__ float sb0[32], ssc0[32], ssh0[32], sb1[32], ssc1[32], ssh1[32], sb2[32];
  __shared__ float sRed[256];
  int tid = threadIdx.x;
  for (int i = tid; i < 32 * CE; i += 256) sW0[i] = W0[i];
  for (int i = tid; i < 1024; i += 256) { sW1[i] = W1[i]; sW2[i] = W2[i]; }
  if (tid < 32) {
    sb0[tid] = b0[tid]; ssc0[tid] = sc0[tid]; ssh0[tid] = sh0[tid];
    sb1[tid] = b1[tid]; ssc1[tid] = sc1[tid]; ssh1[tid] = sh1[tid];
    sb2[tid] = b2[tid];
  }
  __syncthreads();

  int p = blockIdx.x * 256 + tid;       // point id, 0..65535 exactly
  int base = p & ~(NPTS - 1);           // frame start = f*NPTS
  float xi[CIN];
#pragma unroll
  for (int c = 0; c < CIN; ++c) xi[c] = X[(size_t)p * CIN + c];

  float accA[32], accQ[32];
#pragma unroll
  for (int o = 0; o < 32; ++o) {
    accA[o] = (STAGE == 2) ? -__builtin_inff() : 0.f;
    accQ[o] = 0.f;
  }

  for (int k = 0; k < KNN; ++k) {
    int j = nb[(size_t)p * KNN + k];
    const float* Xj = X + (size_t)(base + j) * CIN;
    float ej[CIN];
#pragma unroll
    for (int c = 0; c < CIN; ++c) ej[c] = Xj[c] - xi[c];

    float h[32];
#pragma unroll
    for (int o = 0; o < 32; ++o) {
      float a = sb0[o];
      const float* w = &sW0[o * CE];
#pragma unroll
      for (int c = 0; c < CIN; ++c) a = fmaf(xi[c], w[c], a);
#pragma unroll
      for (int c = 0; c < CIN; ++c) a = fmaf(ej[c], w[CIN + c], a);
      h[o] = a;
    }
    if (STAGE == 0) {
#pragma unroll
      for (int o = 0; o < 32; ++o) { accA[o] += h[o]; accQ[o] += h[o] * h[o]; }
      continue;
    }
#pragma unroll
    for (int o = 0; o < 32; ++o) h[o] = fmaxf(fmaf(h[o], ssc0[o], ssh0[o]), 0.f);

    float t[32];
#pragma unroll
    for (int o = 0; o < 32; ++o) {
      float a = sb1[o];
      const float* w = &sW1[o * 32];
#pragma unroll
      for (int i = 0; i < 32; ++i) a = fmaf(h[i], w[i], a);
      t[o] = a;
    }
    if (STAGE == 1) {
#pragma unroll
      for (int o = 0; o < 32; ++o) { accA[o] += t[o]; accQ[o] += t[o] * t[o]; }
      continue;
    }
#pragma unroll
    for (int o = 0; o < 32; ++o) t[o] = fmaxf(fmaf(t[o], ssc1[o], ssh1[o]), 0.f);
#pragma unroll
    for (int o = 0; o < 32; ++o) {
      float a = sb2[o];
      const float* w = &sW2[o * 32];
#pragma unroll
      for (int i = 0; i < 32; ++i) a = fmaf(t[i], w[i], a);
      accA[o] = fmaxf(accA[o], a);
    }
  }

  if (STAGE == 2) {
#pragma unroll
    for (int o = 0; o < 32; ++o) outFeat[(size_t)p * 32 + o] = accA[o];
  } else {
    // Deterministic intra-block tree reduction, one channel at a time.
#pragma unroll
    for (int o = 0; o < 64; ++o) {
      sRed[tid] = (o < 32) ? accA[o] : accQ[o - 32];
      __syncthreads();
      for (int s = 128; s > 0; s >>= 1) {
        if (tid < s) sRed[tid] += sRed[tid + s];
        __syncthreads();
      }
      if (tid == 0) part[(size_t)blockIdx.x * 64 + o] = sRed[0];
      __syncthreads();
    }
  }
}

// Deterministic cross-block stat reduction + BN scale/shift folding.
__global__ void bn_reduce(const float* __restrict__ part,
                          const float* __restrict__ gamma,
                          const float* __restrict__ beta,
                          float* __restrict__ scale, float* __restrict__ shift) {
  int c = threadIdx.x;
  if (c >= 32) return;
  float S = 0.f, Q = 0.f;
  for (int b = 0; b < 256; ++b) {
    S += part[(size_t)b * 64 + c];
    Q += part[(size_t)b * 64 + 32 + c];
  }
  const float invc = 1.0f / ((float)NFRM * NPTS * KNN);
  float mu  = S * invc;
  float var = Q * invc - mu * mu;
  float r   = rsqrtf(var + 1e-5f);
  float g   = gamma[c] * r;
  scale[c] = g;
  shift[c] = beta[c] - mu * g;
}

// ----------------------- lin1 + max-pool via WMMA --------------------------
__device__ __forceinline__ void atomicMaxF(float* addr, float val) {
  int* ia = (int*)addr;
  int old = *ia;
  while (val > __int_as_float(old)) {
    int prev = atomicCAS(ia, old, __float_as_int(val));
    if (prev == old) break;
    old = prev;
  }
}

__global__ __launch_bounds__(32) void lin1pool_wmma(const float* __restrict__ f0,
                                                    const float* __restrict__ f1,
                                                    const float* __restrict__ f2,
                                                    const float* __restrict__ W,
                                                    const float* __restrict__ b,
                                                    float* __restrict__ pooled) {
  int n0 = blockIdx.x * 16, o0 = blockIdx.y * 16, f = blockIdx.z;
  int l = threadIdx.x, half = l >> 4, m = l & 15;
  size_t rowi = ((size_t)f * NPTS + n0 + m) * 32;

  v8f acc = {};
#pragma unroll
  for (int k0 = 0; k0 < 96; k0 += 4) {
    int ka = k0 + 2 * half;
    const float* src = (k0 < 32) ? f0 : ((k0 < 64) ? f1 : f2);
    int kk = ka & 31;
    v2f a, bb;
    a.x  = src[rowi + kk];
    a.y  = src[rowi + kk + 1];
    bb.x = W[(size_t)(o0 + m) * 96 + ka];
    bb.y = W[(size_t)(o0 + m) * 96 + ka + 1];
    acc = wmma_f32_k4(a, bb, acc);
  }
  // max over the 16 M rows (points) of this tile for each output column.
  float cm = -__builtin_inff();
#pragma unroll
  for (int r = 0; r < 8; ++r) cm = fmaxf(cm, acc[r]);
  cm = fmaxf(cm, __shfl_xor(cm, 16, 32));   // combine M=0..7 with M=8..15
  if (l < 16) atomicMaxF(pooled + (size_t)f * 1024 + o0 + m, cm + b[o0 + m]);
}

__global__ __launch_bounds__(256) void fill_kernel(float* __restrict__ p, float v, int n) {
  int i = blockIdx.x * 256 + threadIdx.x;
  if (i < n) p[i] = v;
}

// ------------------------------ output MLP ---------------------------------
__global__ __launch_bounds__(256) void out_mlp(const float* __restrict__ pooled,
    const float* __restrict__ W0, const float* __restrict__ b0,
    const float* __restrict__ W1, const float* __restrict__ b1,
    const float* __restrict__ W2, const float* __restrict__ b2,
    const float* __restrict__ W3, const float* __restrict__ b3,
    float* __restrict__ out) {
  __shared__ float hA[1024], hB[1024];
  int f = blockIdx.x, tid = threadIdx.x;
  for (int i = tid; i < 1024; i += 256) hA[i] = pooled[(size_t)f * 1024 + i];
  __syncthreads();
  for (int r = 0; r < 4; ++r) {
    int o = tid + 256 * r;
    float a = b0[o];
    const float* w = W0 + (size_t)o * 1024;
    for (int i = 0; i < 1024; ++i) a = fmaf(hA[i], w[i], a);
    hB[o] = fmaxf(a, 0.f);
  }
  __syncthreads();
  {
    int o = tid;
    if (o < 256) {
      float a = b1[o];
      const float* w = W1 + (size_t)o * 1024;
      for (int i = 0; i < 1024; ++i) a = fmaf(hB[i], w[i], a);
      hA[o] = fmaxf(a, 0.f);
    }
  }
  __syncthreads();
  if (tid < 128) {
    float a = b2[tid];
    const float* w = W2 + (size_t)tid * 256;
    for (int i = 0; i < 256; ++i) a = fmaf(hA[i], w[i], a);
    hB[tid] = fmaxf(a, 0.f);
  }
  __syncthreads();
  if (tid < 6) {
    float a = b3[tid];
    const float* w = W3 + (size_t)tid * 128;
    for (int i = 0; i < 128; ++i) a = fmaf(hB[i], w[i], a);
    out[f * 6 + tid] = a;
  }
}

// ------------------------------- driver ------------------------------------
extern "C" void kernel_launch(void* const* d_in, const int* in_sizes, int n_in,
                              void* d_out, int out_size, void* d_ws, size_t ws_size,
                              hipStream_t stream) {
  (void)in_sizes; (void)n_in; (void)out_size; (void)ws_size;
  const float* data = (const float*)d_in[0];
  auto P = [&](int i) { return (const float*)d_in[i]; };

  // workspace layout (floats)
  float* wsf = (float*)d_ws;
  size_t off = 0;
  float* sq = wsf + off;              off += (size_t)NFRM * NPTS;       // 65536
  float* D  = wsf + off;              off += (size_t)NPTS * NPTS;       // 4M (per-frame, L2-resident)
  int*   nb = (int*)(wsf + off);      off += (size_t)NFRM * NPTS * KNN; // 1.31M
  float* feats[3];
  for (int l = 0; l < 3; ++l) { feats[l] = wsf + off; off += (size_t)NFRM * NPTS * 32; }
  float* part   = wsf + off;          off += 256 * 64;
  float* sc0    = wsf + off;          off += 32;
  float* sh0    = wsf + off;          off += 32;
  float* sc1    = wsf + off;          off += 32;
  float* sh1    = wsf + off;          off += 32;
  float* pooled = wsf + off;          off += (size_t)NFRM * 1024;

  for (int l = 0; l < 3; ++l) {
    const float* X = (l == 0) ? data : feats[l - 1];
    int C = (l == 0) ? 9 : 32;

    sqnorm_kernel<<<256, 256, 0, stream>>>(X, sq, C);
    for (int f = 0; f < NFRM; ++f) {
      if (l == 0)
        dist_wmma<9, 12><<<dim3(NPTS / 16, NPTS / 16), 32, 0, stream>>>(X, sq, D, f);
      else
        dist_wmma<32, 32><<<dim3(NPTS / 16, NPTS / 16), 32, 0, stream>>>(X, sq, D, f);
      topk_kernel<<<NPTS, 256, 0, stream>>>(D, nb, f);
    }

    int base = 1 + 10 * l;
    const float *W0 = P(base + 0), *b0 = P(base + 1), *g0 = P(base + 2), *be0 = P(base + 3);
    const float *W1 = P(base + 4), *b1 = P(base + 5), *g1 = P(base + 6), *be1 = P(base + 7);
    const float *W2 = P(base + 8), *b2 = P(base + 9);

    if (l == 0) {
      edge_chain<9, 0><<<256, 256, 0, stream>>>(X, nb, W0, b0, sc0, sh0, W1, b1, sc1, sh1, W2, b2, part, feats[l]);
      bn_reduce<<<1, 32, 0, stream>>>(part, g0, be0, sc0, sh0);
      edge_chain<9, 1><<<256, 256, 0, stream>>>(X, nb, W0, b0, sc0, sh0, W1, b1, sc1, sh1, W2, b2, part, feats[l]);
      bn_reduce<<<1, 32, 0, stream>>>(part, g1, be1, sc1, sh1);
      edge_chain<9, 2><<<256, 256, 0, stream>>>(X, nb, W0, b0, sc0, sh0, W1, b1, sc1, sh1, W2, b2, part, feats[l]);
    } else {
      edge_chain<32, 0><<<256, 256, 0, stream>>>(X, nb, W0, b0, sc0, sh0, W1, b1, sc1, sh1, W2, b2, part, feats[l]);
      bn_reduce<<<1, 32, 0, stream>>>(part, g0, be0, sc0, sh0);
      edge_chain<32, 1><<<256, 256, 0, stream>>>(X, nb, W0, b0, sc0, sh0, W1, b1, sc1, sh1, W2, b2, part, feats[l]);
      bn_reduce<<<1, 32, 0, stream>>>(part, g1, be1, sc1, sh1);
      edge_chain<32, 2><<<256, 256, 0, stream>>>(X, nb, W0, b0, sc0, sh0, W1, b1, sc1, sh1, W2, b2, part, feats[l]);
    }
  }

  fill_kernel<<<(NFRM * 1024 + 255) / 256, 256, 0, stream>>>(pooled, -__builtin_inff(), NFRM * 1024);
  lin1pool_wmma<<<dim3(NPTS / 16, 1024 / 16, NFRM), 32, 0, stream>>>(
      feats[0], feats[1], feats[2], P(31), P(32), pooled);

  out_mlp<<<NFRM, 256, 0, stream>>>(pooled, P(33), P(34), P(35), P(36),
                                    P(37), P(38), P(39), P(40), (float*)d_out);
}